// RFDETRDecoderLayer_13924283974136
// MI455X (gfx1250) — compile-verified
//
#include <hip/hip_runtime.h>
#include <math.h>

#define D_MODEL 256
#define NHEADS  8
#define NPTS    4
#define FFDIM   1024
#define HDIM    32
#define BATCH   8
#define QLEN    900
#define GH      100
#define GW      100
#define HWSZ    (GH*GW)
#define ROWS    (BATCH*QLEN)        // 7200

typedef float v2f __attribute__((ext_vector_type(2)));
typedef float v8f __attribute__((ext_vector_type(8)));

__device__ __forceinline__ v8f wmma4(v2f a, v2f b, v8f c) {
  // D = A(16x4,f32) x B(4x16,f32) + C(16x16,f32)
  return __builtin_amdgcn_wmma_f32_16x16x4_f32(false, a, false, b, (short)0, c, false, false);
}

// ---------------------------------------------------------------------------
// Generic WMMA GEMM:  C[M,N] = act( A[M,K] @ W[N,K]^T + bias[N] )
// One wave computes one 16x16 tile. N multiple of 16, K multiple of 4.
// ---------------------------------------------------------------------------
__global__ void wmma_gemm_bias(const float* __restrict__ A, int lda,
                               const float* __restrict__ W, int ldw,
                               const float* __restrict__ bias,
                               float* __restrict__ C, int ldc,
                               int M, int N, int K, int relu)
{
  int w = blockIdx.x * (blockDim.x >> 5) + (threadIdx.x >> 5);
  int tilesN = N >> 4;
  int tilesM = (M + 15) >> 4;
  if (w >= tilesM * tilesN) return;
  int tm = w / tilesN, tn = w % tilesN;
  int lane = threadIdx.x & 31;
  int half = lane >> 4;          // wave32: lanes 16..31 hold K+2..K+3 (A/B) / rows+8 (C)
  int l16  = lane & 15;

  int rowA = tm * 16 + l16; if (rowA >= M) rowA = M - 1;   // clamp; stores guarded
  int colB = tn * 16 + l16;                                 // N multiple of 16

  const float* Ap = A + (size_t)rowA * lda + half * 2;
  const float* Wp = W + (size_t)colB * ldw + half * 2;

  v8f acc = {0.f,0.f,0.f,0.f,0.f,0.f,0.f,0.f};
  for (int k = 0; k < K; k += 4) {
    v2f a = *(const v2f*)(Ap + k);
    v2f b = *(const v2f*)(Wp + k);
    acc = wmma4(a, b, acc);
  }

  float bv = bias[tn * 16 + l16];
  int colC = tn * 16 + l16;
#pragma unroll
  for (int r = 0; r < 8; ++r) {
    int m = tm * 16 + r + half * 8;
    if (m < M) {
      float v = acc[r] + bv;
      if (relu) v = fmaxf(v, 0.f);
      C[(size_t)m * ldc + colC] = v;
    }
  }
}

// ---------------------------------------------------------------------------
// Flash self-attention. qkv: [B,Q,768] (q|k|v each 256, head h at h*32).
// One wave handles (b, h, 16-query block). Online softmax, P transposed via
// wave-private LDS tile (DS ops are in-order per wave; explicit s_wait_dscnt).
// ---------------------------------------------------------------------------
__global__ void flash_attn_kernel(const float* __restrict__ qkv, float* __restrict__ ctx)
{
  const int nqb = (QLEN + 15) / 16;  // 57
  int w = blockIdx.x * (blockDim.x >> 5) + (threadIdx.x >> 5);
  if (w >= BATCH * NHEADS * nqb) return;
  int qb = w % nqb; int bh = w / nqb;
  int h = bh % NHEADS, b = bh / NHEADS;
  int lane = threadIdx.x & 31, half = lane >> 5 ? 0 : (lane >> 4), l16 = lane & 15;
  half = lane >> 4;

  const float* qb_ = qkv + (size_t)b * QLEN * 768 + h * HDIM;
  const float* kb_ = qb_ + D_MODEL;
  const float* vb_ = qb_ + 2 * D_MODEL;

  int qrow = qb * 16 + l16; if (qrow >= QLEN) qrow = QLEN - 1;
  v2f af[8];
#pragma unroll
  for (int t = 0; t < 8; ++t)
    af[t] = *(const v2f*)(qb_ + (size_t)qrow * 768 + t * 4 + half * 2);

  v8f o0 = {0.f,0.f,0.f,0.f,0.f,0.f,0.f,0.f};
  v8f o1 = {0.f,0.f,0.f,0.f,0.f,0.f,0.f,0.f};
  float mi[8], li[8];
#pragma unroll
  for (int r = 0; r < 8; ++r) { mi[r] = -INFINITY; li[r] = 0.f; }

  __shared__ float psh[8][16 * 17];
  float* ps = psh[threadIdx.x >> 5];
  const float scale = 0.17677669529663687f; // 1/sqrt(32)

  for (int jb = 0; jb < nqb; ++jb) {
    int kcol = jb * 16 + l16;
    int kcr  = kcol < QLEN ? kcol : QLEN - 1;
    const float* krow = kb_ + (size_t)kcr * 768 + half * 2;

    v8f s = {0.f,0.f,0.f,0.f,0.f,0.f,0.f,0.f};
#pragma unroll
    for (int t = 0; t < 8; ++t) {
      v2f bf = *(const v2f*)(krow + t * 4);
      s = wmma4(af[t], bf, s);
    }
    bool cval = kcol < QLEN;

    float pr[8];
#pragma unroll
    for (int r = 0; r < 8; ++r) {
      float sv = cval ? s[r] * scale : -INFINITY;
      float mx = sv;                          // row max across the 16-lane half
      mx = fmaxf(mx, __shfl_xor(mx, 1));
      mx = fmaxf(mx, __shfl_xor(mx, 2));
      mx = fmaxf(mx, __shfl_xor(mx, 4));
      mx = fmaxf(mx, __shfl_xor(mx, 8));
      float mn = fmaxf(mi[r], mx);
      float p  = __expf(sv - mn);
      float rs = p;
      rs += __shfl_xor(rs, 1);
      rs += __shfl_xor(rs, 2);
      rs += __shfl_xor(rs, 4);
      rs += __shfl_xor(rs, 8);
      float corr = __expf(mi[r] - mn);
      li[r] = li[r] * corr + rs;
      mi[r] = mn;
      o0[r] *= corr; o1[r] *= corr;
      pr[r] = p;
    }

    // P tile: C-layout -> LDS -> A-layout (wave-private; DS is in-order per wave)
#pragma unroll
    for (int r = 0; r < 8; ++r) ps[(r + half * 8) * 17 + l16] = pr[r];
    asm volatile("s_wait_dscnt 0" ::: "memory");
    v2f pa[4];
#pragma unroll
    for (int t = 0; t < 4; ++t) {
      int c = t * 4 + half * 2;
      pa[t].x = ps[l16 * 17 + c];
      pa[t].y = ps[l16 * 17 + c + 1];
    }

    // O += P @ V  (K=16 chunk -> 4 K-steps, two 16-wide N tiles)
#pragma unroll
    for (int t = 0; t < 4; ++t) {
      int k0  = jb * 16 + t * 4 + half * 2;
      int k0c = k0 < QLEN ? k0 : QLEN - 1;          // P==0 there anyway
      int k1c = (k0 + 1) < QLEN ? (k0 + 1) : QLEN - 1;
      const float* v0 = vb_ + (size_t)k0c * 768;
      const float* v1 = vb_ + (size_t)k1c * 768;
      v2f b0, b1;
      b0.x = v0[l16];      b0.y = v1[l16];
      b1.x = v0[16 + l16]; b1.y = v1[16 + l16];
      o0 = wmma4(pa[t], b0, o0);
      o1 = wmma4(pa[t], b1, o1);
    }
  }

#pragma unroll
  for (int r = 0; r < 8; ++r) {
    int m = qb * 16 + r + half * 8;
    if (m < QLEN) {
      float inv = 1.f / li[r];
      float* orow = ctx + ((size_t)b * QLEN + m) * D_MODEL + h * HDIM;
      orow[l16]      = o0[r] * inv;
      orow[16 + l16] = o1[r] * inv;
    }
  }
}

// ---------------------------------------------------------------------------
// out = LayerNorm(x + y) * s + b, rows of 256
// ---------------------------------------------------------------------------
__global__ void add_layernorm(const float* __restrict__ x, const float* __restrict__ y,
                              const float* __restrict__ s, const float* __restrict__ bsc,
                              float* __restrict__ out)
{
  __shared__ float red[256];
  int row = blockIdx.x, t = threadIdx.x;
  size_t i = (size_t)row * D_MODEL + t;
  float v = x[i] + y[i];
  red[t] = v; __syncthreads();
  for (int o = 128; o > 0; o >>= 1) { if (t < o) red[t] += red[t + o]; __syncthreads(); }
  float mean = red[0] * (1.f / D_MODEL);
  __syncthreads();
  float d = v - mean;
  red[t] = d * d; __syncthreads();
  for (int o = 128; o > 0; o >>= 1) { if (t < o) red[t] += red[t + o]; __syncthreads(); }
  float var = red[0] * (1.f / D_MODEL);
  out[i] = d * rsqrtf(var + 1e-5f) * s[t] + bsc[t];
}

// ref = sigmoid(t1 @ ref_w^T + ref_b), one thread per (row, comp)
__global__ void ref_kernel(const float* __restrict__ t1, const float* __restrict__ rw,
                           const float* __restrict__ rb, float* __restrict__ ref)
{
  int i = blockIdx.x * blockDim.x + threadIdx.x;
  if (i >= ROWS * 2) return;
  int c = i & 1, row = i >> 1;
  const float* x = t1 + (size_t)row * D_MODEL;
  const float* w = rw + c * D_MODEL;
  float s = rb[c];
  for (int k = 0; k < D_MODEL; ++k) s += x[k] * w[k];
  ref[i] = 1.f / (1.f + __expf(-s));
}

// softmax over the 4 points, in place. One thread per (b,q,h).
__global__ void attw_softmax(float* __restrict__ aw)
{
  int i = blockIdx.x * blockDim.x + threadIdx.x;
  if (i >= ROWS * NHEADS) return;
  float* a = aw + (size_t)i * 4;
  float m = fmaxf(fmaxf(a[0], a[1]), fmaxf(a[2], a[3]));
  float e0 = __expf(a[0] - m), e1 = __expf(a[1] - m);
  float e2 = __expf(a[2] - m), e3 = __expf(a[3] - m);
  float inv = 1.f / (e0 + e1 + e2 + e3);
  a[0] = e0 * inv; a[1] = e1 * inv; a[2] = e2 * inv; a[3] = e3 * inv;
}

// Bilinear deformable sampling. One wave per (b,q,h); lane = head dim.
__global__ void deform_sample(const float* __restrict__ val, const float* __restrict__ ref,
                              const float* __restrict__ offs, const float* __restrict__ attw,
                              float* __restrict__ out)
{
  int w = blockIdx.x * (blockDim.x >> 5) + (threadIdx.x >> 5);
  if (w >= ROWS * NHEADS) return;
  int h = w % NHEADS, bq = w / NHEADS;
  int b = bq / QLEN;
  int lane = threadIdx.x & 31;

  float rx = ref[(size_t)bq * 2 + 0];
  float ry = ref[(size_t)bq * 2 + 1];
  const float* vb = val + (size_t)b * HWSZ * D_MODEL + h * HDIM + lane;

  float acc = 0.f;
  for (int p = 0; p < NPTS; ++p) {
    float ox = offs[(size_t)bq * 64 + h * 8 + p * 2 + 0];
    float oy = offs[(size_t)bq * 64 + h * 8 + p * 2 + 1];
    float aw = attw[(size_t)bq * 32 + h * 4 + p];
    float x = (rx + ox * (1.f / GW)) * GW - 0.5f;
    float y = (ry + oy * (1.f / GH)) * GH - 0.5f;
    float x0f = floorf(x), y0f = floorf(y);
    float lx = x - x0f, ly = y - y0f;
    int x0 = (int)x0f, y0 = (int)y0f;

    float smp = 0.f;
#pragma unroll
    for (int cy = 0; cy < 2; ++cy) {
#pragma unroll
      for (int cx = 0; cx < 2; ++cx) {
        int xi = x0 + cx, yi = y0 + cy;
        bool valid = (xi >= 0) && (xi < GW) && (yi >= 0) && (yi < GH);
        int xc = xi < 0 ? 0 : (xi > GW - 1 ? GW - 1 : xi);
        int yc = yi < 0 ? 0 : (yi > GH - 1 ? GH - 1 : yi);
        float g = vb[(size_t)(yc * GW + xc) * D_MODEL];
        float wx = cx ? lx : (1.f - lx);
        float wy = cy ? ly : (1.f - ly);
        smp += (valid ? g : 0.f) * wx * wy;
      }
    }
    acc += aw * smp;
  }
  out[(size_t)bq * D_MODEL + h * HDIM + lane] = acc;
}

// ---------------------------------------------------------------------------
extern "C" void kernel_launch(void* const* d_in, const int* in_sizes, int n_in,
                              void* d_out, int out_size, void* d_ws, size_t ws_size,
                              hipStream_t stream) {
  const float* tgt        = (const float*)d_in[0];
  const float* memory     = (const float*)d_in[1];
  // d_in[2]=H, d_in[3]=W (ints, fixed at 100 by the reference)
  const float* in_proj_w  = (const float*)d_in[4];
  const float* in_proj_b  = (const float*)d_in[5];
  const float* out_proj_w = (const float*)d_in[6];
  const float* out_proj_b = (const float*)d_in[7];
  const float* norm1_s    = (const float*)d_in[8];
  const float* norm1_b    = (const float*)d_in[9];
  const float* norm2_s    = (const float*)d_in[10];
  const float* norm2_b    = (const float*)d_in[11];
  const float* norm3_s    = (const float*)d_in[12];
  const float* norm3_b    = (const float*)d_in[13];
  const float* vproj_w    = (const float*)d_in[14];
  const float* vproj_b    = (const float*)d_in[15];
  const float* offs_w     = (const float*)d_in[16];
  const float* offs_b     = (const float*)d_in[17];
  const float* attw_w     = (const float*)d_in[18];
  const float* attw_b     = (const float*)d_in[19];
  const float* ref_w      = (const float*)d_in[20];
  const float* ref_b      = (const float*)d_in[21];
  const float* cout_w     = (const float*)d_in[22];
  const float* cout_b     = (const float*)d_in[23];
  const float* ffn_w1     = (const float*)d_in[24];
  const float* ffn_b1     = (const float*)d_in[25];
  const float* ffn_w2     = (const float*)d_in[26];
  const float* ffn_b2     = (const float*)d_in[27];

  float* ws = (float*)d_ws;
  size_t off = 0;
  float* qkv     = ws + off; off += (size_t)ROWS * 768;        // 5.53M
  float* ctx     = ws + off; off += (size_t)ROWS * D_MODEL;    // 1.84M
  float* ptmp    = ws + off; off += (size_t)ROWS * D_MODEL;    // reused 3x
  float* t1      = ws + off; off += (size_t)ROWS * D_MODEL;
  float* t2      = ws + off; off += (size_t)ROWS * D_MODEL;
  float* val     = ws + off; off += (size_t)BATCH * HWSZ * D_MODEL; // 20.5M
  float* refbuf  = ws + off; off += (size_t)ROWS * 2;
  float* offsbuf = ws + off; off += (size_t)ROWS * 64;
  float* attwbuf = ws + off; off += (size_t)ROWS * 32;
  float* dsamp   = ws + off; off += (size_t)ROWS * D_MODEL;
  float* ffn_mid = ws + off; off += (size_t)ROWS * FFDIM;      // total ~173 MB

  auto gemm = [&](const float* A, int lda, const float* W, int ldw, const float* bias,
                  float* C, int ldc, int M, int N, int K, int relu) {
    int tiles = ((M + 15) / 16) * (N / 16);
    wmma_gemm_bias<<<(tiles + 7) / 8, 256, 0, stream>>>(A, lda, W, ldw, bias, C, ldc, M, N, K, relu);
  };

  // 1) QKV projection
  gemm(tgt, D_MODEL, in_proj_w, D_MODEL, in_proj_b, qkv, 768, ROWS, 768, D_MODEL, 0);
  // 2) flash self-attention
  {
    int waves = BATCH * NHEADS * ((QLEN + 15) / 16);
    flash_attn_kernel<<<(waves + 7) / 8, 256, 0, stream>>>(qkv, ctx);
  }
  // 3) out projection + 4) residual LN1
  gemm(ctx, D_MODEL, out_proj_w, D_MODEL, out_proj_b, ptmp, D_MODEL, ROWS, D_MODEL, D_MODEL, 0);
  add_layernorm<<<ROWS, 256, 0, stream>>>(tgt, ptmp, norm1_s, norm1_b, t1);
  // 5) value projection of the memory
  gemm(memory, D_MODEL, vproj_w, D_MODEL, vproj_b, val, D_MODEL, BATCH * HWSZ, D_MODEL, D_MODEL, 0);
  // 6) sampling-offset / 7) attention-weight / 8) reference-point projections
  gemm(t1, D_MODEL, offs_w, D_MODEL, offs_b, offsbuf, 64, ROWS, 64, D_MODEL, 0);
  gemm(t1, D_MODEL, attw_w, D_MODEL, attw_b, attwbuf, 32, ROWS, 32, D_MODEL, 0);
  ref_kernel<<<(ROWS * 2 + 255) / 256, 256, 0, stream>>>(t1, ref_w, ref_b, refbuf);
  attw_softmax<<<(ROWS * NHEADS + 255) / 256, 256, 0, stream>>>(attwbuf);
  // 9) bilinear deformable sampling
  deform_sample<<<(ROWS * NHEADS + 7) / 8, 256, 0, stream>>>(val, refbuf, offsbuf, attwbuf, dsamp);
  // 10) cross-attn output projection + residual LN2
  gemm(dsamp, D_MODEL, cout_w, D_MODEL, cout_b, ptmp, D_MODEL, ROWS, D_MODEL, D_MODEL, 0);
  add_layernorm<<<ROWS, 256, 0, stream>>>(t1, ptmp, norm2_s, norm2_b, t2);
  // 11) FFN
  gemm(t2, D_MODEL, ffn_w1, D_MODEL, ffn_b1, ffn_mid, FFDIM, ROWS, FFDIM, D_MODEL, 1);
  gemm(ffn_mid, FFDIM, ffn_w2, FFDIM, ffn_b2, ptmp, D_MODEL, ROWS, D_MODEL, FFDIM, 0);
  add_layernorm<<<ROWS, 256, 0, stream>>>(t2, ptmp, norm3_s, norm3_b, (float*)d_out);

  (void)in_sizes; (void)n_in; (void)out_size; (void)ws_size;
}